// TimestepPermutationQuantizer_37228776521745
// MI455X (gfx1250) — compile-verified
//
#include <hip/hip_runtime.h>
#include <cstdint>
#include <cstddef>

#define THREADS 256
#define C_DIM   4096
#define GROUP   128
#define ROWS_PER_BLOCK 8
#define F4_PER_THREAD (C_DIM / 4 / THREADS)   // 4 float4 chunks per thread per row

typedef float v4f __attribute__((ext_vector_type(4)));   // native vector for b128 ops

// Issue one 16-byte async copy from global memory at sbase plus voff into LDS at byte address lds_byte. GVS mode:
// 64-bit SGPR base + 32-bit per-lane VGPR offset. Tracked by ASYNCcnt.
#define ASYNC_CP16(lds_byte, voff, sbase)                                   \
    asm volatile("global_load_async_to_lds_b128 %0, %1, %2"                 \
                 :: "v"(lds_byte), "v"(voff), "s"(sbase) : "memory")

__global__ __launch_bounds__(THREADS)
void permquant_rows_kernel(const float* __restrict__ x,
                           const int*   __restrict__ perm,
                           float*       __restrict__ out,
                           int rows_total)
{
    __shared__ __align__(16) float buf[2][C_DIM];   // 32 KB double buffer

    const int tid  = threadIdx.x;
    const int lane = tid & 31;
    const int wave = tid >> 5;
    const int row0 = blockIdx.x * ROWS_PER_BLOCK;

    // The low 32 bits of a flat shared-aperture address are the LDS byte offset.
    const unsigned lds_b0 = (unsigned)(uintptr_t)&buf[0][0];
    const unsigned lds_b1 = (unsigned)(uintptr_t)&buf[1][0];

    // Prologue: async-fill buf[0] with row0.
    if (row0 < rows_total) {
        const float* src = x + (size_t)row0 * C_DIM;
        #pragma unroll
        for (int i = 0; i < F4_PER_THREAD; ++i) {
            const unsigned off = (unsigned)((tid + i * THREADS) * 16);
            ASYNC_CP16(lds_b0 + off, off, src);
        }
    }

    int p = 0;
    for (int r = 0; r < ROWS_PER_BLOCK; ++r) {
        const int row = row0 + r;            // uniform across the block
        if (row >= rows_total) break;

        const bool has_next = (r + 1 < ROWS_PER_BLOCK) && (row + 1 < rows_total);
        if (has_next) {
            // Prefetch next row into the other buffer, then wait only for the
            // current buffer's 4 (oldest) async ops per wave.
            const float* src = x + (size_t)(row + 1) * C_DIM;
            const unsigned lb = p ? lds_b0 : lds_b1;
            #pragma unroll
            for (int i = 0; i < F4_PER_THREAD; ++i) {
                const unsigned off = (unsigned)((tid + i * THREADS) * 16);
                ASYNC_CP16(lb + off, off, src);
            }
            asm volatile("s_wait_asynccnt 0x4" ::: "memory");
        } else {
            asm volatile("s_wait_asynccnt 0x0" ::: "memory");
        }
        __syncthreads();                      // buf[p] fully resident for all waves

        float* cur = buf[p];

        // 32 groups of 128 permuted channels per row; 4 groups per wave,
        // 4 elements per lane per group. Each LDS address is read and written
        // by exactly one thread (perm is a bijection) -> in-place, no races.
        #pragma unroll
        for (int gi = 0; gi < 4; ++gi) {
            const int g = wave * 4 + gi;
            const int j = g * GROUP + lane;

            const int c0 = perm[j];
            const int c1 = perm[j + 32];
            const int c2 = perm[j + 64];
            const int c3 = perm[j + 96];

            const float v0 = cur[c0];
            const float v1 = cur[c1];
            const float v2 = cur[c2];
            const float v3 = cur[c3];

            float mn = fminf(fminf(v0, v1), fminf(v2, v3));
            float mx = fmaxf(fmaxf(v0, v1), fmaxf(v2, v3));
            #pragma unroll
            for (int off = 16; off > 0; off >>= 1) {
                mn = fminf(mn, __shfl_xor(mn, off, 32));
                mx = fmaxf(mx, __shfl_xor(mx, off, 32));
            }

            const float d      = fmaxf(mx - mn, 1e-5f);
            const float scale  = d * (1.0f / 15.0f);   // (max-min)/q_max
            const float rscale = 15.0f / d;            // 1/scale
            const float qb = fminf(fmaxf(rintf(-mn * rscale), 0.0f), 15.0f);

            cur[c0] = (fminf(fmaxf(rintf(v0 * rscale) + qb, 0.0f), 15.0f) - qb) * scale;
            cur[c1] = (fminf(fmaxf(rintf(v1 * rscale) + qb, 0.0f), 15.0f) - qb) * scale;
            cur[c2] = (fminf(fmaxf(rintf(v2 * rscale) + qb, 0.0f), 15.0f) - qb) * scale;
            cur[c3] = (fminf(fmaxf(rintf(v3 * rscale) + qb, 0.0f), 15.0f) - qb) * scale;
        }
        __syncthreads();                      // all scatters visible

        // Coalesced streaming store of the finished row (128-bit NT stores).
        {
            v4f*       dst = (v4f*)(out + (size_t)row * C_DIM);
            const v4f* s4  = (const v4f*)cur;
            #pragma unroll
            for (int i = 0; i < F4_PER_THREAD; ++i) {
                const int k = tid + i * THREADS;
                __builtin_nontemporal_store(s4[k], &dst[k]);
            }
        }
        __syncthreads();   // everyone done reading buf[p] before it is refilled
        p ^= 1;
    }
}

extern "C" void kernel_launch(void* const* d_in, const int* in_sizes, int n_in,
                              void* d_out, int out_size, void* d_ws, size_t ws_size,
                              hipStream_t stream) {
    (void)n_in; (void)d_ws; (void)ws_size; (void)out_size;

    const float* x    = (const float*)d_in[0];
    const int*   perm = (const int*)d_in[1];
    // d_in[2] (inverse_perm) is not needed: scatter back through perm itself.
    float* out = (float*)d_out;

    const int C    = in_sizes[1];              // 4096
    const int rows = in_sizes[0] / C;          // 4*4096 = 16384
    const int grid = (rows + ROWS_PER_BLOCK - 1) / ROWS_PER_BLOCK;

    permquant_rows_kernel<<<grid, THREADS, 0, stream>>>(x, perm, out, rows);
}